// TransformerEncoderLayer_53944789238381
// MI455X (gfx1250) — compile-verified
//
#include <hip/hip_runtime.h>
#include <hip/hip_bf16.h>

// Problem constants (match reference)
#define QDIM 128   // QD == H*HD == 128
#define PDIM 64
#define NHEAD 8
#define HDIM 16

// LDS pitch for K-pair-interleaved W image: 256 floats of payload per pair-row
// + 32-float skew so lanes 16..31 (which read pair-row +1) hit banks 32..63
// while lanes 0..15 hit banks 0..31 -> conflict-free ds_load_b64.
#define WPITCH 288

typedef float v2f __attribute__((ext_vector_type(2)));
typedef float v8f __attribute__((ext_vector_type(8)));

// ---------------------------------------------------------------------------
// Y[r, :] = scale * (X[r, :] @ W) + bias   with X: nrows x 128, W: 128 x 128.
// Pure-f32 WMMA (V_WMMA_F32_16X16X4_F32): exact f32 math, matching the f32
// reference up to FMA association.
// Block = 256 threads = 8 waves. W is staged into LDS in K-pair-interleaved
// form: Wl[p*WPITCH + c*2 + j] = W[2p+j][c], so each WMMA B operand
// (b = {W[k][c], W[k+1][c]}) is a single aligned ds_load_b64 straight into an
// even VGPR pair (no repack moves; pairs of loads merge into ds_load_2addr_b64).
// A-fragments are software-pipelined in 4-K-step chunks (16 VGPRs live) so the
// K-loop has no dependent global loads and no scratch spills.
// NOTE: assumes nrows % 16 == 0 (50000 = 3125*16 holds here).
// ---------------------------------------------------------------------------
__global__ __launch_bounds__(256) void gemm128_wmma(
    const float* __restrict__ X, const float* __restrict__ W,
    const float* __restrict__ bias, float* __restrict__ Y,
    float scale, int nrows)
{
  __shared__ float Wl[64 * WPITCH];        // 73.7 KB of 320 KB WGP LDS
  const int tid = threadIdx.x;

  // Cooperative interleaved fill: 64 pair-rows x 32 col-quads = 2048 tasks.
  // Each task reads float4 from rows 2p and 2p+1, writes 8 interleaved floats.
  #pragma unroll
  for (int it = 0; it < 8; ++it) {
    const int task = it * 256 + tid;       // 0..2047
    const int p    = task >> 5;            // pair-row 0..63
    const int c4   = (task & 31) * 4;      // column quad 0,4,...,124
    const float4 a0 = *(const float4*)&W[(size_t)(2 * p + 0) * 128 + c4];
    const float4 a1 = *(const float4*)&W[(size_t)(2 * p + 1) * 128 + c4];
    float* d = &Wl[p * WPITCH + c4 * 2];
    float4 w0; w0.x = a0.x; w0.y = a1.x; w0.z = a0.y; w0.w = a1.y;
    float4 w1; w1.x = a0.z; w1.y = a1.z; w1.z = a0.w; w1.w = a1.w;
    *(float4*)&d[0] = w0;
    *(float4*)&d[4] = w1;
  }

  const int wave = tid >> 5;
  const int lane = tid & 31;
  const int m    = lane & 15;              // row within A-tile / col within B,C
  const int half = lane >> 4;              // lane group 0..15 vs 16..31
  const int r0   = (blockIdx.x * 8 + wave) * 16;

  // A 16x4 layout per K-step kk: lane holds X[r0+m][4kk+2*half], [4kk+2*half+1].
  const float* Xrow = X + (size_t)(r0 + m) * 128 + 2 * half;

  // Prologue of the A pipeline: first group of 4 K-steps (8 VGPRs).
  float2 a_cur[4];
  if (r0 < nrows) {
    #pragma unroll
    for (int j = 0; j < 4; ++j) a_cur[j] = *(const float2*)&Xrow[j * 4];
  }

  __syncthreads();
  if (r0 >= nrows) return;                 // wave-uniform -> EXEC all-ones for WMMA

  v8f acc[8];
  #pragma unroll
  for (int t = 0; t < 8; ++t)
    #pragma unroll
    for (int g = 0; g < 8; ++g) acc[t][g] = 0.0f;

  // 8 groups x 4 K-steps = K = 128. Prefetch group g+1 while computing group g.
  #pragma unroll
  for (int g = 0; g < 8; ++g) {
    float2 a_next[4];
    if (g < 7) {
      #pragma unroll
      for (int j = 0; j < 4; ++j)
        a_next[j] = *(const float2*)&Xrow[(g + 1) * 16 + j * 4];
    }

    #pragma unroll
    for (int j = 0; j < 4; ++j) {
      const int kk = g * 4 + j;
      v2f a; a.x = a_cur[j].x; a.y = a_cur[j].y;

      // Pair-row for this lane: p = 2*kk + half; column base = m*2.
      const float* Wr = &Wl[(size_t)(2 * kk + half) * WPITCH + m * 2];
      #pragma unroll
      for (int t = 0; t < 8; ++t) {
        // b = { W[4kk+2*half][t*16+m], W[4kk+2*half+1][t*16+m] }: one ds_load_b64
        const v2f b = *(const v2f*)&Wr[t * 32];
        acc[t] = __builtin_amdgcn_wmma_f32_16x16x4_f32(
            /*neg_a=*/false, a, /*neg_b=*/false, b,
            /*c_mod=*/(short)0, acc[t], /*reuse_a=*/false, /*reuse_b=*/false);
      }
    }

    if (g < 7) {
      #pragma unroll
      for (int j = 0; j < 4; ++j) a_cur[j] = a_next[j];  // renamed away by unroll
    }
  }

  // C/D 16x16 f32: c[g] = C[g + 8*half][n], n = lane&15
  #pragma unroll
  for (int t = 0; t < 8; ++t) {
    const int col = t * 16 + m;
    const float b = bias ? bias[col] : 0.0f;
    #pragma unroll
    for (int g = 0; g < 8; ++g) {
      const int row = r0 + g + 8 * half;
      Y[(size_t)row * 128 + col] = acc[t][g] * scale + b;
    }
  }
}

// ---------------------------------------------------------------------------
// Init: m = -inf, den = 0, o = 0
// ---------------------------------------------------------------------------
__global__ void init_kernel(float* __restrict__ m, float* __restrict__ den,
                            float* __restrict__ o, int nm, int no)
{
  const int i = blockIdx.x * blockDim.x + threadIdx.x;
  if (i < nm) { m[i] = -__builtin_inff(); den[i] = 0.0f; }
  if (i < no) o[i] = 0.0f;
}

// ---------------------------------------------------------------------------
// attn[e,h] = <qh[dst,h,:], kh[src,h,:]> + edges[e,:] @ Wb[:,h] + bb[h]
// One thread per edge; qh/kh tables are 25.6 MB each -> L2-resident gathers.
// Wb (64x8) and bb cached in LDS.
// ---------------------------------------------------------------------------
__global__ __launch_bounds__(256) void edge_attn_kernel(
    const float* __restrict__ qh, const float* __restrict__ kh,
    const float* __restrict__ edges, const int* __restrict__ edge_index,
    const float* __restrict__ Wb, const float* __restrict__ bb,
    float* __restrict__ attn, int E)
{
  __shared__ float WbL[PDIM * NHEAD];      // 512 floats
  __shared__ float bbL[NHEAD];
  const int tid = threadIdx.x;
  WbL[tid]        = Wb[tid];               // 256 threads cover 512 in two steps
  WbL[tid + 256]  = Wb[tid + 256];
  if (tid < NHEAD) bbL[tid] = bb[tid];
  __syncthreads();

  const int e = blockIdx.x * 256 + tid;
  if (e >= E) return;

  const int dst = edge_index[2 * e + 0];
  const int src = edge_index[2 * e + 1];
  const float* qp = qh + (size_t)dst * 128;
  const float* kp = kh + (size_t)src * 128;

  float acc[NHEAD];
  #pragma unroll
  for (int h = 0; h < NHEAD; ++h) {
    float a = bbL[h];
    #pragma unroll
    for (int d4 = 0; d4 < 4; ++d4) {
      const float4 qv = *(const float4*)&qp[h * 16 + d4 * 4];
      const float4 kv = *(const float4*)&kp[h * 16 + d4 * 4];
      a += qv.x * kv.x + qv.y * kv.y + qv.z * kv.z + qv.w * kv.w;
    }
    acc[h] = a;
  }

  const float* ep = edges + (size_t)e * PDIM;
  #pragma unroll 4
  for (int j4 = 0; j4 < PDIM / 4; ++j4) {
    const float4 ev = *(const float4*)&ep[j4 * 4];
    #pragma unroll
    for (int h = 0; h < NHEAD; ++h) {      // all lanes same LDS addr -> broadcast
      acc[h] += ev.x * WbL[(j4 * 4 + 0) * NHEAD + h]
              + ev.y * WbL[(j4 * 4 + 1) * NHEAD + h]
              + ev.z * WbL[(j4 * 4 + 2) * NHEAD + h]
              + ev.w * WbL[(j4 * 4 + 3) * NHEAD + h];
    }
  }

  #pragma unroll
  for (int h = 0; h < NHEAD; ++h) attn[(size_t)e * NHEAD + h] = acc[h];
}

// Monotonic float-max via integer atomics (works for all signs, init = -inf):
// positives: signed-int max preserves float order; negatives: unsigned min does.
__device__ __forceinline__ void atomicMaxF(float* addr, float v) {
  if (v >= 0.0f) atomicMax((int*)addr, __float_as_int(v));
  else           atomicMin((unsigned int*)addr, __float_as_uint(v));
}

__global__ void seg_max_kernel(const float* __restrict__ attn,
                               const int* __restrict__ edge_index,
                               float* __restrict__ m, int EH)
{
  const int idx = blockIdx.x * blockDim.x + threadIdx.x;
  if (idx >= EH) return;
  const int e = idx >> 3, h = idx & 7;
  const int dst = edge_index[2 * e];
  atomicMaxF(&m[dst * NHEAD + h], attn[idx]);
}

__global__ void exp_den_kernel(float* __restrict__ attn,
                               const int* __restrict__ edge_index,
                               const float* __restrict__ m,
                               float* __restrict__ den, int EH)
{
  const int idx = blockIdx.x * blockDim.x + threadIdx.x;
  if (idx >= EH) return;
  const int e = idx >> 3, h = idx & 7;
  const int dst = edge_index[2 * e];
  const float ex = expf(attn[idx] - m[dst * NHEAD + h]);
  attn[idx] = ex;                                   // overwrite with unnormalized p
  atomicAdd(&den[dst * NHEAD + h], ex);             // global_atomic_add_f32
}

// o[dst, h, :] += (ex/den) * vh[src, h, :]
__global__ void scatter_kernel(const float* __restrict__ attn,
                               const int* __restrict__ edge_index,
                               const float* __restrict__ den,
                               const float* __restrict__ vh,
                               float* __restrict__ o, int EH)
{
  const int idx = blockIdx.x * blockDim.x + threadIdx.x;
  if (idx >= EH) return;
  const int e = idx >> 3, h = idx & 7;
  const int dst = edge_index[2 * e + 0];
  const int src = edge_index[2 * e + 1];
  const float p = attn[idx] / den[dst * NHEAD + h];
  const float* vp = vh + (size_t)src * 128 + h * 16;
  float* op       = o  + (size_t)dst * 128 + h * 16;
  #pragma unroll
  for (int d4 = 0; d4 < 4; ++d4) {
    const float4 vv = *(const float4*)&vp[d4 * 4];
    atomicAdd(&op[d4 * 4 + 0], p * vv.x);
    atomicAdd(&op[d4 * 4 + 1], p * vv.y);
    atomicAdd(&op[d4 * 4 + 2], p * vv.z);
    atomicAdd(&op[d4 * 4 + 3], p * vv.w);
  }
}

// ---------------------------------------------------------------------------
extern "C" void kernel_launch(void* const* d_in, const int* in_sizes, int n_in,
                              void* d_out, int out_size, void* d_ws, size_t ws_size,
                              hipStream_t stream)
{
  const float* q          = (const float*)d_in[0];
  const float* k          = (const float*)d_in[1];
  const float* v          = (const float*)d_in[2];
  const float* edges      = (const float*)d_in[3];
  const int*   edge_index = (const int*)  d_in[4];
  const float* Wq         = (const float*)d_in[5];
  const float* Wk         = (const float*)d_in[6];
  const float* Wv         = (const float*)d_in[7];
  const float* Wo         = (const float*)d_in[8];
  const float* bo         = (const float*)d_in[9];
  const float* Wb         = (const float*)d_in[10];
  const float* bb         = (const float*)d_in[11];
  float* out = (float*)d_out;

  const int N = in_sizes[0] / QDIM;
  const int E = in_sizes[4] / 2;

  // Workspace carve-up (floats): qh | kh | vh | o | attn(E*8) | m(N*8) | den(N*8)
  float* ws = (float*)d_ws;
  const size_t nodeF = (size_t)N * 128;
  float* qh   = ws;
  float* kh   = qh + nodeF;
  float* vh   = kh + nodeF;
  float* o    = vh + nodeF;
  float* attn = o + nodeF;
  float* m    = attn + (size_t)E * NHEAD;
  float* den  = m + (size_t)N * NHEAD;

  // 1) init m/den/o
  {
    const int n = N * 128;
    init_kernel<<<(n + 255) / 256, 256, 0, stream>>>(m, den, o, N * NHEAD, n);
  }

  // 2) projections (WMMA f32). qh scale = HD^-0.5 = 0.25
  const int rowBlocks = ((N + 15) / 16 + 7) / 8;
  gemm128_wmma<<<rowBlocks, 256, 0, stream>>>(q, Wq, nullptr, qh, 0.25f, N);
  gemm128_wmma<<<rowBlocks, 256, 0, stream>>>(k, Wk, nullptr, kh, 1.0f, N);
  gemm128_wmma<<<rowBlocks, 256, 0, stream>>>(v, Wv, nullptr, vh, 1.0f, N);

  // 3) edge attention scores
  edge_attn_kernel<<<(E + 255) / 256, 256, 0, stream>>>(
      qh, kh, edges, edge_index, Wb, bb, attn, E);

  // 4) segmented softmax over dst + weighted scatter of vh
  const int EH = E * NHEAD;
  seg_max_kernel<<<(EH + 255) / 256, 256, 0, stream>>>(attn, edge_index, m, EH);
  exp_den_kernel<<<(EH + 255) / 256, 256, 0, stream>>>(attn, edge_index, m, den, EH);
  scatter_kernel<<<(EH + 255) / 256, 256, 0, stream>>>(attn, edge_index, den, vh, o, EH);

  // 5) output projection with bias -> d_out
  gemm128_wmma<<<rowBlocks, 256, 0, stream>>>(o, Wo, bo, out, 1.0f, N);
}